// SelfAttention_23398981829348
// MI455X (gfx1250) — compile-verified
//
#include <hip/hip_runtime.h>
#include <hip/hip_bf16.h>
#include <math.h>

// ---------------------------------------------------------------------------
// B=32, S=1, DIM=4096, HQ=32, HKV=8, HD=128, W=4096, kv_len=3001. fp32.
// ~950MB mandatory traffic (~41us @ 23.3TB/s). GEMMs run on
// V_WMMA_F32_16X16X4_F32 with a 4-way K-split per block (LDS reduction) to
// shorten the serial WMMA chain; attention stages V tiles into LDS through
// the gfx1250 async-to-LDS path (ASYNCcnt) with double buffering.
// Wave-uniform control values go through readfirstlane so loops use scalar
// branches instead of EXEC-mask divergence bookkeeping.
// ---------------------------------------------------------------------------
#define BB      32
#define DIMK    4096
#define HQn     32
#define HKVn    8
#define HDn     128
#define Wn      4096
#define KVLEN   3001
#define NEWPOS  3000
#define KVPAD   3008          // KVLEN rounded to 16 (softmax padding, p==0)
#define NTILES  375           // 375 * 8 = 3000 rows handled by tile loop
#define QKVN    6144          // 4096 Q + 1024 K + 1024 V per batch row

typedef __attribute__((ext_vector_type(2))) float v2f;
typedef __attribute__((ext_vector_type(8))) float v8f;
typedef int vi4 __attribute__((vector_size(16)));          // builtin V4i type
typedef __attribute__((address_space(1))) vi4 gvi4;        // global v4i
typedef __attribute__((address_space(3))) vi4 lvi4;        // LDS v4i

#define WAIT_ASYNC_0() asm volatile("s_wait_asynccnt 0x0" ::: "memory")
#define WAIT_ASYNC_1() asm volatile("s_wait_asynccnt 0x1" ::: "memory")

// Async global->LDS 16B copy (ASYNCcnt-tracked on gfx1250), sync fallback.
__device__ __forceinline__ void copy16_to_lds(const float* gsrc, float4* ldst) {
#if __has_builtin(__builtin_amdgcn_global_load_async_to_lds_b128)
    __builtin_amdgcn_global_load_async_to_lds_b128(
        (gvi4*)const_cast<float*>(gsrc), (lvi4*)ldst, 0, 0);
#else
    *ldst = *(const float4*)gsrc;
#endif
}

// ---------------------------------------------------------------------------
// C[32 x 16] tile of  C = A(32xK) * W(NxK)^T + bias.
// 4 waves per block, each wave accumulates a K/4 segment with
// V_WMMA_F32_16X16X4_F32, then the partials are reduced through LDS.
// A 16x4 f32 operand: lanes 0-15 = rows, VGPR0 = K+0 / K+2 (lane>=16),
// VGPR1 = K+1 / K+3.  B 4x16 mirrors that with lanes = columns.
// ---------------------------------------------------------------------------
__global__ __launch_bounds__(128)
void gemm32_wmma_f32(const float* __restrict__ A,     // 32 x K row-major
                     const float* __restrict__ Wm,    // N x K row-major
                     const float* __restrict__ bias,  // N
                     float* __restrict__ out,         // 32 rows, stride ldo
                     int K, int ldo, int coff)
{
    const int ntile = blockIdx.x;
    const int tid   = threadIdx.x;
    const int lane  = tid & 31;
    // wave id is wave-uniform: force it into an SGPR for scalar loop control
    const int wvu   = __builtin_amdgcn_readfirstlane(tid >> 5);   // 0..3
    const int l16   = lane & 15;
    const int khalf = (lane >> 4) * 2;      // 0 or 2

    const int kseg = K >> 2;
    const int kbeg = wvu * kseg;
    const int kend = kbeg + kseg;

    const float* a0 = A  + (size_t)l16 * K + khalf;           // rows 0..15
    const float* a1 = A  + (size_t)(l16 + 16) * K + khalf;    // rows 16..31
    const float* wr = Wm + (size_t)(ntile * 16 + l16) * K + khalf;

    v8f c0 = {};
    v8f c1 = {};
    #pragma unroll 2
    for (int k = kbeg; k < kend; k += 4) {
        v2f av0 = *(const v2f*)(a0 + k);
        v2f av1 = *(const v2f*)(a1 + k);
        v2f bv  = *(const v2f*)(wr + k);
        __builtin_prefetch(wr + k + 512, 0, 1);   // global_prefetch_b8
        c0 = __builtin_amdgcn_wmma_f32_16x16x4_f32(false, av0, false, bv,
                                                   (short)0, c0, false, false);
        c1 = __builtin_amdgcn_wmma_f32_16x16x4_f32(false, av1, false, bv,
                                                   (short)0, c1, false, false);
    }

    // Cross-wave reduction of the two 16x16 accumulators.
    __shared__ float red[4][16][32];
    #pragma unroll
    for (int v = 0; v < 8; ++v) {
        red[wvu][v][lane]     = c0[v];
        red[wvu][v + 8][lane] = c1[v];
    }
    __syncthreads();

    if (tid < 32) {
        const int   n  = ntile * 16 + l16;
        const float bb = bias[n];
        float* o = out + coff + n;
        const int mbase = (lane >> 4) * 8;   // D layout: lane holds column n
        #pragma unroll
        for (int v = 0; v < 8; ++v) {
            float s0 = red[0][v][lane] + red[1][v][lane]
                     + red[2][v][lane] + red[3][v][lane];
            float s1 = red[0][v + 8][lane] + red[1][v + 8][lane]
                     + red[2][v + 8][lane] + red[3][v + 8][lane];
            o[(size_t)(mbase + v)      * ldo] = s0 + bb;
            o[(size_t)(mbase + v + 16) * ldo] = s1 + bb;
        }
    }
}

// ---------------------------------------------------------------------------
// Attention: one block (256 thr = 8 waves) per (b, kv_head).  K/V are read
// once and shared across the 4 grouped q heads (4x traffic saving).
// ---------------------------------------------------------------------------
__global__ __launch_bounds__(256)
void attn_kernel(const float* __restrict__ qkv,       // B x 6144
                 const float* __restrict__ fcos,      // 64
                 const float* __restrict__ fsin,      // 64
                 const float* __restrict__ cache_k,   // B x W x HKV x HD
                 const float* __restrict__ cache_v,
                 float* __restrict__ attn_out)        // B x (HQ*HD)
{
    const int kvh = blockIdx.x;     // 0..7
    const int b   = blockIdx.y;     // 0..31
    const int tid  = threadIdx.x;
    const int lane = tid & 31;
    const int wvu  = __builtin_amdgcn_readfirstlane(tid >> 5);  // wave id 0..7

    __shared__ float qs[4 * HDn];            // RoPE'd q, 4 heads
    __shared__ float kn[HDn];                // RoPE'd new key
    __shared__ float vn[HDn];                // new value
    __shared__ float4 sc4[KVPAD];            // scores / probs (4 heads per key)
    __shared__ float4 vt4[2][8 * 32];        // double-buffered 8-row V tile
    __shared__ float wred[8][4], wsum[8][4], mred[4], invs[4];

    const size_t qbase = (size_t)b * QKVN;

    // ---- load q (4 heads) + new k with RoPE, new v ----
    {
        int r = tid >> 6, i = tid & 63;                 // 4*64 rotation pairs
        const float* qh = qkv + qbase + (size_t)(kvh * 4 + r) * HDn;
        float c = fcos[i], s = fsin[i];
        float x0 = qh[2 * i], x1 = qh[2 * i + 1];
        qs[r * HDn + 2 * i]     = x0 * c - x1 * s;
        qs[r * HDn + 2 * i + 1] = x0 * s + x1 * c;
        if (tid < 64) {
            const float* kh = qkv + qbase + 4096 + (size_t)kvh * HDn;
            float k0 = kh[2 * i], k1 = kh[2 * i + 1];
            kn[2 * i]     = k0 * c - k1 * s;
            kn[2 * i + 1] = k0 * s + k1 * c;
        }
        if (tid >= 64 && tid < 192) {
            int d = tid - 64;
            vn[d] = qkv[qbase + 5120 + (size_t)kvh * HDn + d];
        }
    }
    __syncthreads();

    const float scale = 1.0f / sqrtf((float)HDn);
    const float4* qs4 = (const float4*)qs;
    const float4* kn4 = (const float4*)kn;
    const float4* vn4 = (const float4*)vn;

    // ---- pass 1: scores, wave-per-key (coalesced 512B per key) ----
    float lm0 = -INFINITY, lm1 = -INFINITY, lm2 = -INFINITY, lm3 = -INFINITY;
    for (int j = wvu; j < KVPAD; j += 8) {      // scalar loop (wave-uniform)
        if (j > NEWPOS) {
            if (lane == 0) sc4[j] = make_float4(-INFINITY, -INFINITY, -INFINITY, -INFINITY);
            continue;
        }
        float4 kk;
        if (j == NEWPOS) {
            kk = kn4[lane];
        } else {
            const float* kp = cache_k + ((size_t)b * Wn + j) * (HKVn * HDn) + (size_t)kvh * HDn;
            kk = ((const float4*)kp)[lane];
        }
        float p0, p1, p2, p3;
        {
            float4 q0 = qs4[0 * 32 + lane], q1 = qs4[1 * 32 + lane];
            float4 q2 = qs4[2 * 32 + lane], q3 = qs4[3 * 32 + lane];
            p0 = kk.x * q0.x + kk.y * q0.y + kk.z * q0.z + kk.w * q0.w;
            p1 = kk.x * q1.x + kk.y * q1.y + kk.z * q1.z + kk.w * q1.w;
            p2 = kk.x * q2.x + kk.y * q2.y + kk.z * q2.z + kk.w * q2.w;
            p3 = kk.x * q3.x + kk.y * q3.y + kk.z * q3.z + kk.w * q3.w;
        }
        #pragma unroll
        for (int off = 16; off >= 1; off >>= 1) {
            p0 += __shfl_xor(p0, off, 32);
            p1 += __shfl_xor(p1, off, 32);
            p2 += __shfl_xor(p2, off, 32);
            p3 += __shfl_xor(p3, off, 32);
        }
        p0 *= scale; p1 *= scale; p2 *= scale; p3 *= scale;
        lm0 = fmaxf(lm0, p0); lm1 = fmaxf(lm1, p1);
        lm2 = fmaxf(lm2, p2); lm3 = fmaxf(lm3, p3);
        if (lane == 0) sc4[j] = make_float4(p0, p1, p2, p3);
    }
    if (lane == 0) { wred[wvu][0] = lm0; wred[wvu][1] = lm1; wred[wvu][2] = lm2; wred[wvu][3] = lm3; }
    __syncthreads();
    if (tid < 4) {
        float m = -INFINITY;
        #pragma unroll
        for (int w = 0; w < 8; ++w) m = fmaxf(m, wred[w][tid]);
        mred[tid] = m;
    }
    __syncthreads();

    // ---- pass 2: exp + sum ----
    float ls0 = 0.f, ls1 = 0.f, ls2 = 0.f, ls3 = 0.f;
    {
        float m0 = mred[0], m1 = mred[1], m2 = mred[2], m3 = mred[3];
        for (int j = tid; j < KVPAD; j += 256) {
            float4 sv = sc4[j];
            float4 pv;
            pv.x = __expf(sv.x - m0); pv.y = __expf(sv.y - m1);
            pv.z = __expf(sv.z - m2); pv.w = __expf(sv.w - m3);
            sc4[j] = pv;
            ls0 += pv.x; ls1 += pv.y; ls2 += pv.z; ls3 += pv.w;
        }
    }
    #pragma unroll
    for (int off = 16; off >= 1; off >>= 1) {
        ls0 += __shfl_xor(ls0, off, 32);
        ls1 += __shfl_xor(ls1, off, 32);
        ls2 += __shfl_xor(ls2, off, 32);
        ls3 += __shfl_xor(ls3, off, 32);
    }
    if (lane == 0) { wsum[wvu][0] = ls0; wsum[wvu][1] = ls1; wsum[wvu][2] = ls2; wsum[wvu][3] = ls3; }
    __syncthreads();
    if (tid < 4) {
        float s = 0.f;
        #pragma unroll
        for (int w = 0; w < 8; ++w) s += wsum[w][tid];
        invs[tid] = 1.0f / s;
    }
    __syncthreads();

    // ---- pass 3: P*V, async double-buffered 8-row V tiles ----
    // Tiles cover exactly j in [0,3000): every thread issues exactly one
    // async b128 per tile, so ASYNCcnt thresholds are uniform.
    const int d4   = tid & 31;                                        // per-lane
    const int r    = __builtin_amdgcn_readfirstlane((tid >> 5) & 3);  // wave-uniform
    const int half = __builtin_amdgcn_readfirstlane(tid >> 7);        // wave-uniform
    const int lrow = wvu;               // loader: row 0..7
    const float* scf = (const float*)sc4;
    const float* vbase = cache_v + (size_t)b * Wn * (HKVn * HDn) + (size_t)kvh * HDn;

    // prime tile 0
    copy16_to_lds(vbase + (size_t)lrow * (HKVn * HDn) + d4 * 4, &vt4[0][lrow * 32 + d4]);

    float ax = 0.f, ay = 0.f, az = 0.f, aw = 0.f;
    for (int t = 0; t < NTILES; ++t) {
        if (t + 1 < NTILES) {
            int jn = (t + 1) * 8 + lrow;
            copy16_to_lds(vbase + (size_t)jn * (HKVn * HDn) + d4 * 4,
                          &vt4[(t + 1) & 1][lrow * 32 + d4]);
            WAIT_ASYNC_1();             // the older (tile t) loads completed
        } else {
            WAIT_ASYNC_0();
        }
        __syncthreads();                // tile t visible to all waves
        const float4* vt = vt4[t & 1];
        #pragma unroll
        for (int rr = 0; rr < 4; ++rr) {
            int row = half * 4 + rr;
            float  p  = scf[(t * 8 + row) * 4 + r];
            float4 vv = vt[row * 32 + d4];
            ax += p * vv.x; ay += p * vv.y; az += p * vv.z; aw += p * vv.w;
        }
        __syncthreads();                // done reading before buffer reuse
    }

    // new-token value term (only one row-half adds it)
    if (half == 0) {
        float p = scf[NEWPOS * 4 + r];
        float4 vv = vn4[d4];
        ax += p * vv.x; ay += p * vv.y; az += p * vv.z; aw += p * vv.w;
    }

    // combine the two row-halves through LDS (partner is in another wave)
    __syncthreads();
    ((float4*)vt4)[tid] = make_float4(ax, ay, az, aw);
    __syncthreads();
    if (tid < 128) {
        float4 a = ((float4*)vt4)[tid], c = ((float4*)vt4)[tid + 128];
        float iv = invs[r];
        float4 o;
        o.x = (a.x + c.x) * iv; o.y = (a.y + c.y) * iv;
        o.z = (a.z + c.z) * iv; o.w = (a.w + c.w) * iv;
        float* op = attn_out + (size_t)b * (HQn * HDn)
                  + (size_t)(kvh * 4 + r) * HDn + d4 * 4;
        *(float4*)op = o;
    }
}

// ---------------------------------------------------------------------------
extern "C" void kernel_launch(void* const* d_in, const int* in_sizes, int n_in,
                              void* d_out, int out_size, void* d_ws, size_t ws_size,
                              hipStream_t stream) {
    const float* x     = (const float*)d_in[0];
    const float* fcos  = (const float*)d_in[1];
    const float* fsin  = (const float*)d_in[2];
    const float* wq_w  = (const float*)d_in[3];
    const float* wq_b  = (const float*)d_in[4];
    const float* wk_w  = (const float*)d_in[5];
    const float* wk_b  = (const float*)d_in[6];
    const float* wv_w  = (const float*)d_in[7];
    const float* wv_b  = (const float*)d_in[8];
    const float* wo_w  = (const float*)d_in[9];
    const float* wo_b  = (const float*)d_in[10];
    const float* cache_k = (const float*)d_in[11];
    const float* cache_v = (const float*)d_in[12];

    float* qkv  = (float*)d_ws;                 // 32 x 6144
    float* attn = qkv + (size_t)BB * QKVN;      // 32 x 4096

    // Q / K / V projections (fused qkv buffer, row stride 6144)
    gemm32_wmma_f32<<<DIMK / 16, 128, 0, stream>>>(x, wq_w, wq_b, qkv, DIMK, QKVN, 0);
    gemm32_wmma_f32<<<(HKVn * HDn) / 16, 128, 0, stream>>>(x, wk_w, wk_b, qkv, DIMK, QKVN, 4096);
    gemm32_wmma_f32<<<(HKVn * HDn) / 16, 128, 0, stream>>>(x, wv_w, wv_b, qkv, DIMK, QKVN, 5120);

    // GQA decode attention: one block per (b, kv_head)
    attn_kernel<<<dim3(HKVn, BB), 256, 0, stream>>>(qkv, fcos, fsin, cache_k, cache_v, attn);

    // Output projection straight into d_out
    gemm32_wmma_f32<<<DIMK / 16, 128, 0, stream>>>(attn, wo_w, wo_b, (float*)d_out, DIMK, DIMK, 0);
}